// Attention_20212116095191
// MI455X (gfx1250) — compile-verified
//
#include <hip/hip_runtime.h>
#include <math.h>

#define BB   8
#define LQ   128
#define LM   512
#define QS   512
#define MS   512
#define HS   256

typedef __attribute__((ext_vector_type(2))) float v2f;
typedef __attribute__((ext_vector_type(8))) float v8f;

#define GEMM_WAVES 4

// ---------------------------------------------------------------------------
// Batched GEMM  C[z] = A[z] @ W[z] (+ bias), f32, via V_WMMA_F32_16X16X4_F32.
// Compile-time N,K so B-matrix strides fold into immediate offsets.
// One wave owns a 32x16 C tile: two accumulators share each B fragment
// (2 x b64 A-loads + 2 x b32 B-loads per 2 WMMAs).
// block = (32, GEMM_WAVES), grid = (M/32, N/(16*GEMM_WAVES), batch).
// Lane layout (ISA 7.12.2): A: row m = lane&15, VGPR i holds K = 2*(lane>>4)+i.
// B mirrored on N.  C/D: row = v + 8*(lane>>4).
// ---------------------------------------------------------------------------
template <int N, int K, bool HAS_BIAS>
__global__ __launch_bounds__(32 * GEMM_WAVES)
void wmma_gemm_f32(const float* __restrict__ A, const float* __restrict__ W,
                   const float* __restrict__ bias, float* __restrict__ C,
                   long sA, long sW, long sC)
{
    const int lane = threadIdx.x;                       // 0..31
    const int m0   = blockIdx.x * 32;                   // two 16-row tiles
    const int n0   = (blockIdx.y * GEMM_WAVES + threadIdx.y) * 16;
    const int z    = blockIdx.z;
    A += (long)z * sA;  W += (long)z * sW;  C += (long)z * sC;

    const int mr = lane & 15;
    const int kk = (lane >> 4) << 1;                    // 0 or 2
    const int n  = lane & 15;

    const float* arow0 = A + (long)(m0 + mr) * K + kk;  // 8B aligned (kk even, K even)
    const float* arow1 = arow0 + (long)16 * K;
    const float* bcol  = W + (long)kk * N + (n0 + n);

    v8f acc0 = {0.f, 0.f, 0.f, 0.f, 0.f, 0.f, 0.f, 0.f};
    v8f acc1 = {0.f, 0.f, 0.f, 0.f, 0.f, 0.f, 0.f, 0.f};

#pragma unroll 4
    for (int k = 0; k < K; k += 4) {
        v2f a0 = *(const v2f*)(arow0 + k);              // A[m   ][k+kk .. +1]
        v2f a1 = *(const v2f*)(arow1 + k);              // A[m+16][k+kk .. +1]
        v2f b;
        b.x = bcol[k * N];                              // W[k+kk  ][n] (const stride)
        b.y = bcol[k * N + N];                          // W[k+kk+1][n]
        acc0 = __builtin_amdgcn_wmma_f32_16x16x4_f32(
                   false, a0, false, b, (short)0, acc0, false, false);
        acc1 = __builtin_amdgcn_wmma_f32_16x16x4_f32(
                   false, a1, false, b, (short)0, acc1, false, false);
    }

    const float bv    = HAS_BIAS ? bias[n0 + n] : 0.f;
    const int   mbase = m0 + ((lane >> 4) << 3);
    float* c0 = C + (long)mbase * N + (n0 + n);
#pragma unroll
    for (int r = 0; r < 8; ++r)  c0[r * N]        = acc0[r] + bv;
#pragma unroll
    for (int r = 0; r < 8; ++r)  c0[(16 + r) * N] = acc1[r] + bv;
}

// ---------------------------------------------------------------------------
// tanh(x) = 1 - 2 / (exp(2x) + 1): 2 transcendentals (v_exp_f32, v_rcp_f32)
// + 3 VALU ops. Exact at +/-inf, NaN-propagating, ~2 ulp.
// ---------------------------------------------------------------------------
__device__ __forceinline__ float fast_tanh(float x)
{
    float e = __expf(2.0f * x);
    return 1.0f - 2.0f * __builtin_amdgcn_rcpf(e + 1.0f);
}

// ---------------------------------------------------------------------------
// Fused Bahdanau score + mask + softmax for one (b,q):
//   attn[m] = sum_h v[h] * tanh(qp[b,q,h] + kp[b,m,h]);  mask -> -1e24;
//   weights[b,q,m] = softmax_m(attn).
// 512 threads (16 waves): thread == memory slot m. qp/v staged in LDS and
// read back as ds_load_b128; kp rows streamed from L2 as float4 (each lane
// walks a contiguous 1 KB row -> high L0 line reuse).
// ---------------------------------------------------------------------------
__global__ __launch_bounds__(512)
void attn_softmax_kernel(const float* __restrict__ qp, const float* __restrict__ kp,
                         const float* __restrict__ vvec,
                         const unsigned char* __restrict__ mask,
                         float* __restrict__ weights)
{
    const int b = blockIdx.y, q = blockIdx.x;
    const int tid = threadIdx.x;                        // == m, 0..511

    __shared__ float s_q[HS];
    __shared__ float s_v[HS];
    __shared__ float red[512];

    if (tid < HS) {
        s_q[tid] = qp[((long)b * LQ + q) * HS + tid];
        s_v[tid] = vvec[tid];
    }
    __syncthreads();

    const float4* krow4 = (const float4*)(kp + ((long)b * LM + tid) * HS);
    const float4* q4    = (const float4*)s_q;
    const float4* v4    = (const float4*)s_v;

    float s = 0.f;
#pragma unroll 4
    for (int i = 0; i < HS / 4; ++i) {
        float4 kv = krow4[i];
        float4 qv = q4[i];
        float4 vv = v4[i];
        s += vv.x * fast_tanh(qv.x + kv.x);
        s += vv.y * fast_tanh(qv.y + kv.y);
        s += vv.z * fast_tanh(qv.z + kv.z);
        s += vv.w * fast_tanh(qv.w + kv.w);
    }
    if (mask[(long)b * LM + tid]) s = -1e24f;

    red[tid] = s; __syncthreads();
#pragma unroll
    for (int off = 256; off > 0; off >>= 1) {
        if (tid < off) red[tid] = fmaxf(red[tid], red[tid + off]);
        __syncthreads();
    }
    const float mx = red[0];
    __syncthreads();

    const float e = __expf(s - mx);
    red[tid] = e; __syncthreads();
#pragma unroll
    for (int off = 256; off > 0; off >>= 1) {
        if (tid < off) red[tid] += red[tid + off];
        __syncthreads();
    }
    const float sum = red[0];

    weights[((long)b * LQ + q) * LM + tid] = e / sum;
}

// ---------------------------------------------------------------------------
extern "C" void kernel_launch(void* const* d_in, const int* in_sizes, int n_in,
                              void* d_out, int out_size, void* d_ws, size_t ws_size,
                              hipStream_t stream)
{
    const float*         query  = (const float*)d_in[0];          // [B,LQ,QS]
    const float*         memory = (const float*)d_in[1];          // [B,LM,MS]
    const unsigned char* mask   = (const unsigned char*)d_in[2];  // [B,LM] bool
    const float*         Wq     = (const float*)d_in[3];          // [QS,HS]
    const float*         bq     = (const float*)d_in[4];          // [HS]
    const float*         Wm     = (const float*)d_in[5];          // [MS,HS]
    const float*         v      = (const float*)d_in[6];          // [HS]

    float* out      = (float*)d_out;
    float* weighted = out;                                // [B,LQ,MS]
    float* weights  = out + (long)BB * LQ * MS;           // [B,LQ,LM]

    float* qp = (float*)d_ws;                             // [B*LQ, HS] = 1 MB
    float* kp = qp + (long)BB * LQ * HS;                  // [B*LM, HS] = 4 MB

    dim3 blk(32, GEMM_WAVES);

    // 1) q_proj = query @ Wq + bq    ([1024 x 512] @ [512 x 256])
    wmma_gemm_f32<HS, QS, true>
        <<<dim3((BB * LQ) / 32, HS / (16 * GEMM_WAVES), 1), blk, 0, stream>>>(
            query, Wq, bq, qp, 0, 0, 0);

    // 2) k_proj = memory @ Wm       ([4096 x 512] @ [512 x 256])
    wmma_gemm_f32<HS, MS, false>
        <<<dim3((BB * LM) / 32, HS / (16 * GEMM_WAVES), 1), blk, 0, stream>>>(
            memory, Wm, nullptr, kp, 0, 0, 0);

    // 3) fused tanh-score + mask + softmax -> weights
    attn_softmax_kernel<<<dim3(LQ, BB), 512, 0, stream>>>(qp, kp, v, mask, weights);

    // 4) weighted = weights @ memory, batched over B ([128 x 512] @ [512 x 512])
    wmma_gemm_f32<MS, LM, false>
        <<<dim3(LQ / 32, MS / (16 * GEMM_WAVES), BB), blk, 0, stream>>>(
            weights, memory, nullptr, weighted,
            (long)LQ * LM, (long)LM * MS, (long)LQ * MS);
}